// CaptionModel_53188874994240
// MI455X (gfx1250) — compile-verified
//
#include <hip/hip_runtime.h>
#include <hip/hip_bf16.h>

// ---------------------------------------------------------------------------
// Beam-search step, B=128 beams, V=128000 vocab, K=128, wave32 (gfx1250).
// Stage 1: per-row exact top-128 via single-pass streaming selection with a
//          2048-slot LDS candidate buffer + bitonic compaction. Unique 64-bit
//          keys (value<<32)|~index give jax.lax.top_k semantics exactly
//          (value desc, index asc on ties). One read of the 65.5MB matrix.
// Stage 2: same selector over the 16384 candidate sums (flat-index ties).
// Stage 3: sequence/logprob/sum writes + state gathers; states staged through
//          LDS with the gfx1250 async DMA path (ASYNCcnt).
// ---------------------------------------------------------------------------

#if defined(__has_builtin)
#if __has_builtin(__builtin_amdgcn_global_load_async_to_lds_b128) && \
    __has_builtin(__builtin_amdgcn_global_store_async_from_lds_b128) && \
    __has_builtin(__builtin_amdgcn_s_wait_asynccnt)
#define HAVE_ASYNC_LDS 1
#endif
#endif

#define NT 256        // threads per selection block
#define KSEL 128      // top-k size (beam_size)
#define CAPP 2048     // LDS candidate buffer (power of two, sort domain)
#define WATERMARK 512 // compact when count reaches this at a chunk boundary
#define CHUNK (NT * 4)

// 16-byte vector type matching the async-LDS builtin signatures
typedef int v4i __attribute__((vector_size(4 * sizeof(int))));
typedef __attribute__((address_space(1))) v4i* gv4i_ptr; // global (AS1)
typedef __attribute__((address_space(3))) v4i* lv4i_ptr; // LDS (AS3)

// Order-preserving float -> u32 map (total order, larger float -> larger key)
__device__ __forceinline__ unsigned fmap(float f) {
    unsigned u = __float_as_uint(f);
    return (u & 0x80000000u) ? ~u : (u | 0x80000000u);
}
__device__ __forceinline__ float funmap(unsigned k) {
    unsigned u = (k & 0x80000000u) ? (k & 0x7FFFFFFFu) : ~k;
    return __uint_as_float(u);
}

struct StreamShared {
    unsigned long long buf[CAPP];
    unsigned long long thresh;
    unsigned count;
};

struct RowLoader {
    const float* row;
    int V;
    __device__ void load4(int j0, float f[4]) const {
        if (j0 + 3 < V) {
            float4 v = *reinterpret_cast<const float4*>(row + j0);
            f[0] = v.x; f[1] = v.y; f[2] = v.z; f[3] = v.w;
        } else {
            for (int e = 0; e < 4; ++e)
                f[e] = (j0 + e < V) ? row[j0 + e] : 0.0f;
        }
        // EOS penalty on last vocab entry (applied every step before top_k)
        if ((V - 1) >= j0 && (V - 1) <= j0 + 3) f[(V - 1) - j0] += -1000.0f;
    }
    __device__ void prefetch(int j) const {
        if (j < V) __builtin_prefetch(row + j, 0, 0); // global_prefetch
    }
};

struct CandLoader {
    const float* ys;   // [128][128] per-row sorted top values (flat)
    const float* bls;  // beam_logprob_sum [128]
    int N;
    __device__ void load4(int j0, float f[4]) const {
        const float s = bls[j0 >> 7]; // quad never crosses a row (j0 % 4 == 0)
        if (j0 + 3 < N) {
            float4 v = *reinterpret_cast<const float4*>(ys + j0);
            f[0] = s + v.x; f[1] = s + v.y; f[2] = s + v.z; f[3] = s + v.w;
        } else {
            for (int e = 0; e < 4; ++e)
                f[e] = (j0 + e < N) ? s + ys[j0 + e] : 0.0f;
        }
    }
    __device__ void prefetch(int) const {}
};

// In-LDS bitonic sort of CAPP u64 keys, descending. NT threads.
__device__ __forceinline__ void sort_desc_2048(unsigned long long* buf) {
    const unsigned tid = threadIdx.x;
    for (unsigned k = 2; k <= CAPP; k <<= 1) {
        for (unsigned j = k >> 1; j > 0; j >>= 1) {
            __syncthreads();
            for (unsigned e = tid; e < CAPP / 2; e += NT) {
                const unsigned i = 2u * e - (e & (j - 1u)); // bit j clear
                const unsigned p = i + j;
                unsigned long long a = buf[i];
                unsigned long long b = buf[p];
                const bool desc = ((i & k) == 0u);
                if (desc ? (a < b) : (a > b)) { buf[i] = b; buf[p] = a; }
            }
        }
    }
    __syncthreads();
}

// Exact top-128 of the N keys produced by ld; on return sh.buf[0..127] holds
// them sorted descending by (value, then ascending index). NT threads.
template <typename Loader>
__device__ void stream_topk128(StreamShared& sh, int N, Loader ld) {
    const unsigned tid = threadIdx.x;
    for (int i = tid; i < CAPP; i += NT) sh.buf[i] = 0ull; // -inf sentinels
    if (tid == 0) { sh.thresh = 0ull; sh.count = 0u; }
    __syncthreads();

    for (int base = 0; base < N; base += CHUNK) {
        const unsigned long long th = sh.thresh;
        const int j0 = base + (int)tid * 4;
        if (j0 < N) {
            if ((tid & 7u) == 0u) ld.prefetch(j0 + 2 * CHUNK);
            float f[4];
            ld.load4(j0, f);
#pragma unroll
            for (int e = 0; e < 4; ++e) {
                const int j = j0 + e;
                if (j < N) {
                    const unsigned long long comb =
                        ((unsigned long long)fmap(f[e]) << 32) |
                        (unsigned)(~(unsigned)j);
                    if (comb > th) {
                        unsigned slot = atomicAdd(&sh.count, 1u);
                        if (slot < CAPP) sh.buf[slot] = comb;
                    }
                }
            }
        }
        __syncthreads();
        const bool last = (base + CHUNK >= N);
        if (sh.count >= WATERMARK || last) {
            sort_desc_2048(sh.buf); // barriers inside; ends with barrier
            if (tid == 0) {
                sh.count = KSEL;            // keep top-128, rest is overwritable
                sh.thresh = sh.buf[KSEL - 1];
            }
        }
        __syncthreads();
    }
}

// ---- Stage 1: per-row top-128 of logprobs (one block per beam row) -------
__global__ void topk_rows_kernel(const float* __restrict__ logprobsf,
                                 const int* __restrict__ t_ptr,
                                 float* __restrict__ ys_ws,
                                 int* __restrict__ ix_ws, int V) {
    __shared__ StreamShared sh;
    const int t = *t_ptr;
    const int b = blockIdx.x;
    if (t < 1 && b > 0) return; // rows = 1 when t < 1
    RowLoader ld{logprobsf + (size_t)b * V, V};
    stream_topk128(sh, V, ld);
    const unsigned tid = threadIdx.x;
    if (tid < KSEL) {
        const unsigned long long c = sh.buf[tid];
        const unsigned key = (unsigned)(c >> 32);
        const int idx = (int)(~(unsigned)c);
        ys_ws[b * KSEL + tid] = funmap(key);
        ix_ws[b * KSEL + tid] = idx;
    }
}

// ---- Stage 2: global top-128 over candidates (single block) --------------
__global__ void topk_global_kernel(const float* __restrict__ ys_ws,
                                   const int* __restrict__ ix_ws,
                                   const float* __restrict__ bls,
                                   const int* __restrict__ t_ptr,
                                   int* __restrict__ ws_q,
                                   int* __restrict__ ws_tok,
                                   float* __restrict__ ws_loc,
                                   float* __restrict__ ws_topp) {
    __shared__ StreamShared sh;
    const int t = *t_ptr;
    const int rows = (t >= 1) ? KSEL : 1;
    const int N = rows * KSEL;
    CandLoader ld{ys_ws, bls, N};
    stream_topk128(sh, N, ld);
    const unsigned tid = threadIdx.x;
    if (tid < KSEL) {
        const unsigned long long c = sh.buf[tid];
        const unsigned key = (unsigned)(c >> 32);
        const unsigned flat = ~(unsigned)c;
        ws_q[tid] = (int)(flat >> 7);      // q_sel = flat // 128
        ws_tok[tid] = ix_ws[flat];         // ix[q, c]
        ws_loc[tid] = ys_ws[flat];         // ys[q, c]
        ws_topp[tid] = funmap(key);        // bls[q] + ys[q, c]
    }
}

// ---- Stage 3a: write new_seq, new_logps, new_sum -------------------------
__global__ void write_seq_kernel(const int* __restrict__ beam_seq,
                                 const float* __restrict__ beam_logps,
                                 const int* __restrict__ t_ptr,
                                 const int* __restrict__ ws_q,
                                 const int* __restrict__ ws_tok,
                                 const float* __restrict__ ws_loc,
                                 const float* __restrict__ ws_topp,
                                 float* __restrict__ out, int T, int B) {
    const int idx = blockIdx.x * blockDim.x + threadIdx.x;
    const int t = *t_ptr;
    const int S = T * B;
    if (idx < S) {
        const int r = idx / B, i = idx % B;
        float v;
        if (r < t)       v = (float)beam_seq[r * B + ws_q[i]];
        else if (r == t) v = (float)ws_tok[i];
        else             v = (float)beam_seq[idx];
        out[idx] = v;
    } else if (idx < 2 * S) {
        const int k = idx - S;
        const int r = k / B, i = k % B;
        float v;
        if (r < t)       v = beam_logps[r * B + ws_q[i]];
        else if (r == t) v = ws_loc[i];
        else             v = beam_logps[k];
        out[idx] = v;
    } else if (idx < 2 * S + B) {
        out[idx] = ws_topp[idx - 2 * S];
    }
}

// ---- Stage 3b: state gather via async global->LDS->global DMA path -------
// grid = 2*L*B blocks (h then c), 256 threads, H = 1024 floats per block.
__global__ void gather_state_kernel(const float* __restrict__ state_h,
                                    const float* __restrict__ state_c,
                                    const int* __restrict__ ws_q,
                                    float* __restrict__ out, int L, int B,
                                    int H) {
    const int LB = L * B;
    const int blk = blockIdx.x;
    const int which = blk / LB;
    const int rem = blk % LB;
    const int l = rem / B, i = rem % B;
    const int q = ws_q[i];
    const float* src =
        (which ? state_c : state_h) + ((size_t)l * B + q) * H;
    float* dst = out + (size_t)which * (size_t)LB * H + ((size_t)l * B + i) * H;

    __align__(16) __shared__ float buf[1024]; // H floats, 4 KB

#if defined(HAVE_ASYNC_LDS) && defined(__AMDGCN__)
    // Each of 256 lanes moves 16B: 4KB tile staged through LDS using the
    // gfx1250 async-copy path (tracked by ASYNCcnt).
    {
        const unsigned tid = threadIdx.x;
        gv4i_ptr gsrc = (gv4i_ptr)(void*)((char*)(void*)src + 16u * tid);
        gv4i_ptr gdst = (gv4i_ptr)(void*)((char*)(void*)dst + 16u * tid);
        lv4i_ptr lb   = (lv4i_ptr)(void*)((char*)(void*)buf + 16u * tid);
        __builtin_amdgcn_global_load_async_to_lds_b128(gsrc, lb, 0, 0);
        __builtin_amdgcn_s_wait_asynccnt(0);
        __syncthreads();
        __builtin_amdgcn_global_store_async_from_lds_b128(gdst, lb, 0, 0);
        __builtin_amdgcn_s_wait_asynccnt(0);
    }
#else
    {
        const unsigned tid = threadIdx.x;
        float4 v = reinterpret_cast<const float4*>(src)[tid];
        buf[tid] = v.x; // keep buf live; trivial
        reinterpret_cast<float4*>(dst)[tid] = v;
    }
#endif
}

// ---------------------------------------------------------------------------
extern "C" void kernel_launch(void* const* d_in, const int* in_sizes, int n_in,
                              void* d_out, int out_size, void* d_ws,
                              size_t ws_size, hipStream_t stream) {
    const float* logprobsf = (const float*)d_in[0];
    const int* beam_seq = (const int*)d_in[1];
    const float* beam_logps = (const float*)d_in[2];
    const float* bls = (const float*)d_in[3];
    const float* state_h = (const float*)d_in[4];
    const float* state_c = (const float*)d_in[5];
    const int* t_ptr = (const int*)d_in[6];

    const int B = in_sizes[3];            // 128 beams
    const int V = in_sizes[0] / B;        // 128000 vocab
    const int T = in_sizes[1] / B;        // 128 timesteps
    const int H = 1024;                   // hidden
    const int L = in_sizes[4] / (B * H);  // 2 layers

    float* out = (float*)d_out;
    char* ws = (char*)d_ws;
    float* ys_ws = (float*)ws;                          // 16384 f32
    int* ix_ws = (int*)(ws + 65536);                    // 16384 i32
    int* ws_q = (int*)(ws + 131072);                    // 128 i32
    int* ws_tok = (int*)(ws + 131072 + 512);            // 128 i32
    float* ws_loc = (float*)(ws + 131072 + 1024);       // 128 f32
    float* ws_topp = (float*)(ws + 131072 + 1536);      // 128 f32

    // Stage 1: per-row exact top-128, one streaming read of logprobs
    topk_rows_kernel<<<B, NT, 0, stream>>>(logprobsf, t_ptr, ys_ws, ix_ws, V);

    // Stage 2: global exact top-128 over candidates
    topk_global_kernel<<<1, NT, 0, stream>>>(ys_ws, ix_ws, bls, t_ptr, ws_q,
                                             ws_tok, ws_loc, ws_topp);

    // Stage 3a: sequences / logprobs / sums
    const int S = T * B;
    const int n1 = 2 * S + B;
    write_seq_kernel<<<(n1 + NT - 1) / NT, NT, 0, stream>>>(
        beam_seq, beam_logps, t_ptr, ws_q, ws_tok, ws_loc, ws_topp, out, T, B);

    // Stage 3b: state gathers (async LDS DMA path)
    gather_state_kernel<<<2 * L * B, NT, 0, stream>>>(
        state_h, state_c, ws_q, out + 2 * (size_t)S + B, L, B, H);
}